// GraphNeuralNetwork_10574209483258
// MI455X (gfx1250) — compile-verified
//
#include <hip/hip_runtime.h>

#define B_ 32
#define N_ 1024
#define E_ 8192
#define H_ 64
#define SDIM_ 2053
#define ENC_NEG_INF 0x007FFFFFu

typedef __attribute__((ext_vector_type(2))) float v2f;
typedef __attribute__((ext_vector_type(8))) float v8f;

// Monotone float->uint encoding so unsigned atomicMax == float max.
__device__ __forceinline__ unsigned fenc(float f) {
    unsigned u = __float_as_uint(f);
    return u ^ ((u & 0x80000000u) ? 0xFFFFFFFFu : 0x80000000u);
}
__device__ __forceinline__ float fdec(unsigned u) {
    unsigned bits = (u & 0x80000000u) ? (u ^ 0x80000000u) : ~u;
    return __uint_as_float(bits);
}

// One wave computes a 16-row x (NT*16)-col tile.
// A rows in LDS (row stride rstride floats), W^T in LDS as [NT*16][KDIM+4].
// A-fragment layout (f32 16x16x4): lanes 0-15 hold M=lane,K=k..k+1;
// lanes 16-31 hold M=lane-16,K=k+2..k+3.  B mirrors with N=lane&15.
template<int KDIM, int NT>
__device__ __forceinline__ void gemm16(const float* rows, int rstride, int arow,
                                       const float* wT, int lane, v8f* acc) {
    const int m  = lane & 15;
    const int kh = (lane & 16) ? 2 : 0;
    const float* ap = rows + (arow + m) * rstride + kh;
#pragma unroll
    for (int k = 0; k < KDIM; k += 4) {
        v2f a = *(const v2f*)(ap + k);
#pragma unroll
        for (int t = 0; t < NT; ++t) {
            v2f b = *(const v2f*)(wT + (t * 16 + m) * (KDIM + 4) + kh + k);
            acc[t] = __builtin_amdgcn_wmma_f32_16x16x4_f32(
                false, a, false, b, (short)0, acc[t], false, false);
        }
    }
}

// Write C tiles (+bias, optional relu) back to the wave's LDS rows.
// C/D layout: VGPR v, lanes 0-15: M=v, N=lane; lanes 16-31: M=v+8, N=lane-16.
template<int NT, bool RELU>
__device__ __forceinline__ void store_lds(float* rows, int rstride, int arow,
                                          const v8f* acc, const float* bf, int lane) {
    const int m  = lane & 15;
    const int hi = (lane & 16) ? 8 : 0;
#pragma unroll
    for (int t = 0; t < NT; ++t) {
#pragma unroll
        for (int v = 0; v < 8; ++v) {
            float x = acc[t][v] + bf[t];
            if (RELU) x = fmaxf(x, 0.0f);
            rows[(arow + v + hi) * rstride + t * 16 + m] = x;
        }
    }
}

// ---------------- encoder: build node_input + inputFunction MLP ----------------
__global__ __launch_bounds__(256) void k_input(const float* __restrict__ state,
                                               const float* __restrict__ node_feat,
                                               const float* __restrict__ Wi1,
                                               const float* __restrict__ bi1,
                                               const float* __restrict__ Wi2,
                                               const float* __restrict__ bi2,
                                               float* __restrict__ node_in_g,
                                               float* __restrict__ s_out) {
    extern __shared__ float sm[];
    float* w1t  = sm;               // 64*20
    float* w2t  = w1t + 64 * 20;    // 64*68
    float* rows = w2t + 64 * 68;    // 128*84
    const int tid = threadIdx.x;
    const int blockbase = blockIdx.x * 128;

    for (int idx = tid; idx < 64 * 20; idx += 256) {
        int n = idx / 20, kk = idx % 20;
        w1t[idx] = (kk < 13) ? Wi1[kk * 64 + n] : 0.0f;
    }
    for (int idx = tid; idx < 64 * 68; idx += 256) {
        int n = idx / 68, kk = idx % 68;
        w2t[idx] = (kk < 64) ? Wi2[kk * 64 + n] : 0.0f;
    }
    for (int idx = tid; idx < 128 * 16; idx += 256) {
        int row = idx >> 4, c = idx & 15;
        int r = blockbase + row;
        int n = r >> 5, b = r & 31;
        float v = 0.0f;
        if (c < 5)       v = state[b * SDIM_ + c];
        else if (c == 5) v = state[b * SDIM_ + 5 + n];
        else if (c == 6) v = state[b * SDIM_ + 5 + N_ + n];
        else if (c < 13) v = node_feat[n * 6 + (c - 7)];
        rows[row * 84 + c] = v;
        node_in_g[r * 16 + c] = v;
    }
    __syncthreads();

    const int lane = tid & 31, m = lane & 15;
    const int arow = (tid >> 5) * 16;
    float b1f[4], b2f[4];
#pragma unroll
    for (int t = 0; t < 4; ++t) { b1f[t] = bi1[t * 16 + m]; b2f[t] = bi2[t * 16 + m]; }

    v8f acc[4] = {};
    gemm16<16, 4>(rows, 84, arow, w1t, lane, acc);
    store_lds<4, true>(rows, 84, arow, acc, b1f, lane);
    v8f acc2[4] = {};
    gemm16<64, 4>(rows, 84, arow, w2t, lane, acc2);

    const int hi = (lane & 16) ? 8 : 0;
#pragma unroll
    for (int t = 0; t < 4; ++t)
#pragma unroll
        for (int v = 0; v < 8; ++v) {
            int r = blockbase + arow + v + hi;
            s_out[r * 64 + t * 16 + m] = acc2[t][v] + b2f[t];
        }
}

// ---------------- m_hat init ----------------
__global__ __launch_bounds__(256) void k_fill(uint4* __restrict__ p) {
    int i = blockIdx.x * 256 + threadIdx.x;
    uint4 v; v.x = v.y = v.z = v.w = ENC_NEG_INF;
    p[i] = v;
}

// ---------------- edge message MLP + fused segment-max scatter ----------------
__global__ __launch_bounds__(256) void k_msg(const float* __restrict__ s_cur,
                                             const float* __restrict__ node_in_g,
                                             const float* __restrict__ edge_feat,
                                             const int* __restrict__ src,
                                             const int* __restrict__ dst,
                                             const float* __restrict__ Wm1,
                                             const float* __restrict__ bm1,
                                             const float* __restrict__ Wm2,
                                             const float* __restrict__ bm2,
                                             unsigned* __restrict__ mhat) {
    extern __shared__ float sm[];
    float* w1t  = sm;               // 64*84
    float* w2t  = w1t + 64 * 84;    // 64*68
    float* rows = w2t + 64 * 68;    // 128*84
    const int tid = threadIdx.x;
    const int blockbase = blockIdx.x * 128;   // 4 edges x 32 batch per block

    for (int idx = tid; idx < 64 * 84; idx += 256) {
        int n = idx / 84, kk = idx % 84;
        w1t[idx] = (kk < 78) ? Wm1[kk * 64 + n] : 0.0f;
    }
    for (int idx = tid; idx < 64 * 68; idx += 256) {
        int n = idx / 68, kk = idx % 68;
        w2t[idx] = (kk < 64) ? Wm2[kk * 64 + n] : 0.0f;
    }
    for (int idx = tid; idx < 128 * 80; idx += 256) {
        int row = idx / 80, c = idx % 80;
        int r = blockbase + row;
        int e = r >> 5, b = r & 31;
        int sr = src[e] * 32 + b;
        float v;
        if (c < 64)       v = s_cur[sr * 64 + c];
        else if (c == 64) v = edge_feat[e];
        else if (c < 78)  v = node_in_g[sr * 16 + (c - 65)];
        else              v = 0.0f;
        rows[row * 84 + c] = v;
    }
    __syncthreads();

    const int lane = tid & 31, m = lane & 15;
    const int arow = (tid >> 5) * 16;
    float b1f[4], b2f[4];
#pragma unroll
    for (int t = 0; t < 4; ++t) { b1f[t] = bm1[t * 16 + m]; b2f[t] = bm2[t * 16 + m]; }

    v8f acc[4] = {};
    gemm16<80, 4>(rows, 84, arow, w1t, lane, acc);
    store_lds<4, true>(rows, 84, arow, acc, b1f, lane);
    v8f acc2[4] = {};
    gemm16<64, 4>(rows, 84, arow, w2t, lane, acc2);

    // all 16 rows of this wave belong to one edge -> scalar dst
    const int e  = (blockbase + arow) >> 5;
    const int dn = dst[e];
    const int hi = (lane & 16) ? 8 : 0;
#pragma unroll
    for (int t = 0; t < 4; ++t)
#pragma unroll
        for (int v = 0; v < 8; ++v) {
            int r = blockbase + arow + v + hi;
            int b = r & 31;
            float x = acc2[t][v] + b2f[t];
            atomicMax(&mhat[(dn * 32 + b) * 64 + t * 16 + m], fenc(x));
        }
}

// ---------------- update MLP ----------------
__global__ __launch_bounds__(256) void k_update(const unsigned* __restrict__ mhat,
                                                const float* __restrict__ s_cur,
                                                const float* __restrict__ Wu1,
                                                const float* __restrict__ bu1,
                                                const float* __restrict__ Wu2,
                                                const float* __restrict__ bu2,
                                                float* __restrict__ s_out) {
    extern __shared__ float sm[];
    float* w1t  = sm;                // 64*132
    float* w2t  = w1t + 64 * 132;    // 64*68
    float* rows = w2t + 64 * 68;     // 128*132
    const int tid = threadIdx.x;
    const int blockbase = blockIdx.x * 128;

    for (int idx = tid; idx < 64 * 132; idx += 256) {
        int n = idx / 132, kk = idx % 132;
        w1t[idx] = (kk < 128) ? Wu1[kk * 64 + n] : 0.0f;
    }
    for (int idx = tid; idx < 64 * 68; idx += 256) {
        int n = idx / 68, kk = idx % 68;
        w2t[idx] = (kk < 64) ? Wu2[kk * 64 + n] : 0.0f;
    }
    for (int idx = tid; idx < 128 * 128; idx += 256) {
        int row = idx >> 7, c = idx & 127;
        int r = blockbase + row;
        float v;
        if (c < 64) {
            unsigned u = mhat[r * 64 + c];
            v = (u == ENC_NEG_INF) ? 0.0f : fdec(u);   // empty segment -> 0
        } else {
            v = s_cur[r * 64 + (c - 64)];
        }
        rows[row * 132 + c] = v;
    }
    __syncthreads();

    const int lane = tid & 31, m = lane & 15;
    const int arow = (tid >> 5) * 16;
    float b1f[4], b2f[4];
#pragma unroll
    for (int t = 0; t < 4; ++t) { b1f[t] = bu1[t * 16 + m]; b2f[t] = bu2[t * 16 + m]; }

    v8f acc[4] = {};
    gemm16<128, 4>(rows, 132, arow, w1t, lane, acc);
    store_lds<4, true>(rows, 132, arow, acc, b1f, lane);
    v8f acc2[4] = {};
    gemm16<64, 4>(rows, 132, arow, w2t, lane, acc2);

    const int hi = (lane & 16) ? 8 : 0;
#pragma unroll
    for (int t = 0; t < 4; ++t)
#pragma unroll
        for (int v = 0; v < 8; ++v) {
            int r = blockbase + arow + v + hi;
            s_out[r * 64 + t * 16 + m] = acc2[t][v] + b2f[t];
        }
}

// ---------------- output MLP + L2 normalize ----------------
__global__ __launch_bounds__(256) void k_output(const float* __restrict__ s_cur,
                                                const float* __restrict__ Wo1,
                                                const float* __restrict__ bo1,
                                                const float* __restrict__ Wo2,
                                                const float* __restrict__ bo2,
                                                float* __restrict__ out) {
    extern __shared__ float sm[];
    float* w1t  = sm;               // 64*68
    float* w2t  = w1t + 64 * 68;    // 32*68
    float* rows = w2t + 32 * 68;    // 128*68
    const int tid = threadIdx.x;
    const int blockbase = blockIdx.x * 128;

    for (int idx = tid; idx < 64 * 68; idx += 256) {
        int n = idx / 68, kk = idx % 68;
        w1t[idx] = (kk < 64) ? Wo1[kk * 64 + n] : 0.0f;
    }
    for (int idx = tid; idx < 32 * 68; idx += 256) {
        int n = idx / 68, kk = idx % 68;
        w2t[idx] = (kk < 64) ? Wo2[kk * 32 + n] : 0.0f;   // Wo2 is [64][32]
    }
    for (int idx = tid; idx < 128 * 64; idx += 256) {
        int row = idx >> 6, c = idx & 63;
        rows[row * 68 + c] = s_cur[(blockbase + row) * 64 + c];
    }
    __syncthreads();

    const int lane = tid & 31, m = lane & 15;
    const int arow = (tid >> 5) * 16;
    float b1f[4], b2f[2];
#pragma unroll
    for (int t = 0; t < 4; ++t) b1f[t] = bo1[t * 16 + m];
#pragma unroll
    for (int t = 0; t < 2; ++t) b2f[t] = bo2[t * 16 + m];

    v8f acc[4] = {};
    gemm16<64, 4>(rows, 68, arow, w1t, lane, acc);
    store_lds<4, true>(rows, 68, arow, acc, b1f, lane);
    v8f acc2[2] = {};
    gemm16<64, 2>(rows, 68, arow, w2t, lane, acc2);
    store_lds<2, false>(rows, 68, arow, acc2, b2f, lane);
    __syncthreads();

    if (lane < 16) {
        int row = arow + lane;
        float ss = 0.0f;
#pragma unroll
        for (int c = 0; c < 32; ++c) {
            float x = rows[row * 68 + c];
            ss += x * x;
        }
        float sc = 1.0f / fmaxf(sqrtf(ss), 1e-12f);
        int r = blockbase + row;
#pragma unroll
        for (int c = 0; c < 32; ++c) out[r * 32 + c] = rows[row * 68 + c] * sc;
    }
}

extern "C" void kernel_launch(void* const* d_in, const int* in_sizes, int n_in,
                              void* d_out, int out_size, void* d_ws, size_t ws_size,
                              hipStream_t stream) {
    const float* state     = (const float*)d_in[0];
    const float* node_feat = (const float*)d_in[1];
    const float* edge_feat = (const float*)d_in[2];
    const int*   src       = (const int*)d_in[3];
    const int*   dst       = (const int*)d_in[4];
    const float* Wi1 = (const float*)d_in[5];  const float* bi1 = (const float*)d_in[6];
    const float* Wi2 = (const float*)d_in[7];  const float* bi2 = (const float*)d_in[8];
    const float* Wm1 = (const float*)d_in[9];  const float* bm1 = (const float*)d_in[10];
    const float* Wm2 = (const float*)d_in[11]; const float* bm2 = (const float*)d_in[12];
    const float* Wu1 = (const float*)d_in[13]; const float* bu1 = (const float*)d_in[14];
    const float* Wu2 = (const float*)d_in[15]; const float* bu2 = (const float*)d_in[16];
    const float* Wo1 = (const float*)d_in[17]; const float* bo1 = (const float*)d_in[18];
    const float* Wo2 = (const float*)d_in[19]; const float* bo2 = (const float*)d_in[20];

    const int R = N_ * B_;                     // 32768 node-batch rows
    float* ws        = (float*)d_ws;
    float* node_in_g = ws;                     // R*16
    float* s_a       = node_in_g + (size_t)R * 16;
    float* s_b       = s_a + (size_t)R * 64;
    unsigned* mhat   = (unsigned*)(s_b + (size_t)R * 64);

    const size_t smA = (size_t)(64 * 20 + 64 * 68 + 128 * 84) * sizeof(float);
    const size_t smB = (size_t)(64 * 84 + 64 * 68 + 128 * 84) * sizeof(float);
    const size_t smC = (size_t)(64 * 132 + 64 * 68 + 128 * 132) * sizeof(float);
    const size_t smD = (size_t)(64 * 68 + 32 * 68 + 128 * 68) * sizeof(float);

    k_input<<<R / 128, 256, smA, stream>>>(state, node_feat, Wi1, bi1, Wi2, bi2,
                                           node_in_g, s_a);
    float* cur = s_a;
    float* nxt = s_b;
    for (int it = 0; it < 6; ++it) {
        k_fill<<<(R * 64 / 4) / 256, 256, 0, stream>>>((uint4*)mhat);
        k_msg<<<(E_ * B_) / 128, 256, smB, stream>>>(cur, node_in_g, edge_feat, src, dst,
                                                     Wm1, bm1, Wm2, bm2, mhat);
        k_update<<<R / 128, 256, smC, stream>>>(mhat, cur, Wu1, bu1, Wu2, bu2, nxt);
        float* t = cur; cur = nxt; nxt = t;
    }
    k_output<<<R / 128, 256, smD, stream>>>(cur, Wo1, bo1, Wo2, bo2, (float*)d_out);
}